// TabularEmbedding_48026324304120
// MI455X (gfx1250) — compile-verified
//
#include <hip/hip_runtime.h>
#include <math.h>

// Problem constants (match reference)
#define N_B 2048
#define N_T 13
#define N_D 157
#define N_H 16
#define NCAT 11
#define NCONT 146
#define MAXV 7

#define THREADS 256
#define ROWS_PER_BLOCK N_T                 // 13 rows per block (one b index)
#define F4_PER_ROW ((N_D * N_H) / 4)       // 628
#define Q_PER_BLOCK (ROWS_PER_BLOCK * F4_PER_ROW)  // 8164

// ---- CDNA5 async global->LDS path (guarded so compile never fails) ----
#if defined(__has_builtin)
#  if __has_builtin(__builtin_amdgcn_global_load_async_to_lds_b128) && \
      __has_builtin(__builtin_amdgcn_global_load_async_to_lds_b32)  && \
      __has_builtin(__builtin_amdgcn_s_wait_asynccnt)
#    define USE_ASYNC 1
#  endif
#endif

// Builtin signatures (from hipcc diagnostics):
//   b128: (AS1 v4i*, AS3 v4i*, imm offset, imm cpol)
//   b32 : (AS1 int*, AS3 int*, imm offset, imm cpol)
typedef int v4i __attribute__((__vector_size__(4 * sizeof(int))));
#define G128(p) ((__attribute__((address_space(1))) v4i*)(p))
#define L128(p) ((__attribute__((address_space(3))) v4i*)(p))
#define G32(p)  ((__attribute__((address_space(1))) int*)(p))
#define L32(p)  ((__attribute__((address_space(3))) int*)(p))

__global__ __launch_bounds__(THREADS) void tab_emb_gelu_kernel(
    const float* __restrict__ x,          // (B*T, D)
    const float* __restrict__ cat_table,  // (11, 7, 16)
    const float* __restrict__ W,          // (146, 16)
    const float* __restrict__ b,          // (146, 16)
    const float* __restrict__ na_emb,     // (16)
    const float* __restrict__ pos_emb,    // (157, 16) flat
    float* __restrict__ out)              // (B*T, 2512)
{
    __shared__ float  s_pos[N_D * N_H];            // 2512, linear (read at i)
    __shared__ float  s_na[N_H];                   // 16
    __shared__ float  s_x[ROWS_PER_BLOCK * N_D];   // 2041
    __shared__ float  s_cat[N_H * NCAT * MAXV];    // transposed: [h][d*7+v]
    __shared__ float2 s_wb[N_H * NCONT];           // transposed: [h][c] = (W,b)

    const int tid = threadIdx.x;
    const int blk = blockIdx.x;
    const float* xg = x + (size_t)blk * (ROWS_PER_BLOCK * N_D);

#ifdef USE_ASYNC
    // pos_emb: 628 x b128 async copies (16B-aligned source & dest)
    for (int t = tid; t < F4_PER_ROW; t += THREADS)
        __builtin_amdgcn_global_load_async_to_lds_b128(
            G128(pos_emb + t * 4), L128(s_pos + t * 4), 0, 0);
    // x rows for this block: base offset is only 4B-aligned -> b32 async
    for (int t = tid; t < ROWS_PER_BLOCK * N_D; t += THREADS)
        __builtin_amdgcn_global_load_async_to_lds_b32(
            G32(xg + t), L32(s_x + t), 0, 0);
    if (tid < N_H)
        __builtin_amdgcn_global_load_async_to_lds_b32(
            G32(na_emb + tid), L32(s_na + tid), 0, 0);
#else
    for (int t = tid; t < N_D * N_H; t += THREADS) s_pos[t] = pos_emb[t];
    for (int t = tid; t < ROWS_PER_BLOCK * N_D; t += THREADS) s_x[t] = xg[t];
    if (tid < N_H) s_na[tid] = na_emb[tid];
#endif

    // Stage W/b/cat transposed so the compute loop (lanes stride d2 by 4)
    // hits LDS with stride-1 addresses instead of 32-way bank conflicts.
    for (int t = tid; t < NCONT * N_H; t += THREADS) {
        int c = t >> 4, h = t & 15;                // source is (c, h) row-major
        s_wb[h * NCONT + c] = make_float2(W[t], b[t]);
    }
    for (int t = tid; t < NCAT * MAXV * N_H; t += THREADS) {
        int p = t >> 4, h = t & 15;                // p = d*7 + v
        s_cat[h * (NCAT * MAXV) + p] = cat_table[t];
    }

#ifdef USE_ASYNC
    __builtin_amdgcn_s_wait_asynccnt(0);
#endif
    __syncthreads();

    // Each thread produces one float4 of output per iteration (coalesced b128).
    float4* out4 = (float4*)out + (size_t)blk * Q_PER_BLOCK;
    for (int q = tid; q < Q_PER_BLOCK; q += THREADS) {
        unsigned row = (unsigned)q / 628u;
        unsigned j   = (unsigned)q - row * 628u;
        unsigned i0  = j * 4u;                 // flat index within row, i = d_out*16+h_out
        unsigned h2  = i0 / 157u;              // source emb coords: value at i comes
        unsigned d2  = i0 - h2 * 157u;         // from emb[d2 = i%157, h2 = i/157]
        const float4 pos4 = *(const float4*)(s_pos + i0);
        const float pv[4] = {pos4.x, pos4.y, pos4.z, pos4.w};
        const float* xr = s_x + row * N_D;
        float r[4];
#pragma unroll
        for (int k = 0; k < 4; ++k) {
            float xv = xr[d2];
            bool nn = !(xv == xv);             // isnan
            float xc = nn ? 0.0f : xv;
            float v;
            if (d2 < NCAT) {
                int idx = (int)xc;             // truncate, matches astype(int32)
                v = s_cat[h2 * (NCAT * MAXV) + d2 * MAXV + idx];
            } else {
                float2 wb = s_wb[h2 * NCONT + (d2 - NCAT)];
                v = fmaf(xc, wb.x, wb.y);
            }
            if (nn) v = s_na[h2];
            float u = v + pv[k];
            // exact GELU: 0.5*u*(1+erf(u/sqrt(2)))
            r[k] = 0.5f * u * (1.0f + erff(u * 0.7071067811865475f));
            if (++d2 == (unsigned)N_D) { d2 = 0u; ++h2; }
        }
        out4[q] = make_float4(r[0], r[1], r[2], r[3]);
    }
}

extern "C" void kernel_launch(void* const* d_in, const int* in_sizes, int n_in,
                              void* d_out, int out_size, void* d_ws, size_t ws_size,
                              hipStream_t stream) {
    const float* x   = (const float*)d_in[0];
    const float* cat = (const float*)d_in[1];
    const float* W   = (const float*)d_in[2];
    const float* b   = (const float*)d_in[3];
    const float* na  = (const float*)d_in[4];
    const float* pos = (const float*)d_in[5];
    float* out = (float*)d_out;
    tab_emb_gelu_kernel<<<dim3(N_B), dim3(THREADS), 0, stream>>>(
        x, cat, W, b, na, pos, out);
}